// QuantumNet_20650202759720
// MI455X (gfx1250) — compile-verified
//
#include <hip/hip_runtime.h>
#include <math.h>

// ---------------------------------------------------------------------------
// Quantum kernel target alignment on gfx1250 (MI455X).
// 10 qubits (DIM=1024), 64 points, 4096 pairs. One wave32 per pair; complex
// state as re/im f32 planes in LDS. 4-qubit gate groups fused into real 16x16
// matrices applied with V_WMMA_F32_16X16X4_F32 (f32 matches reference
// complex64 precision). Qubits 8,9 via 4x4 VALU apply. All wave-uniform
// operands live in LDS; WMMA fragments are preloaded in batches and the
// re/im accumulation chains are interleaved to hide WMMA/LDS latency.
// ---------------------------------------------------------------------------

typedef __attribute__((ext_vector_type(2))) float v2f;
typedef __attribute__((ext_vector_type(8))) float v8f;

#define NQ    10
#define QDIM  1024
#define NPTS  64
#define DFEAT 10
#define NLAY  5

__device__ __forceinline__ v8f wmma4(v2f a, v2f b, v8f c) {
  // D = A(16x4,f32) * B(4x16,f32) + C(16x16,f32)
  return __builtin_amdgcn_wmma_f32_16x16x4_f32(
      /*neg_a=*/false, a, /*neg_b=*/false, b,
      /*c_mod=*/(short)0, c, /*reuse_a=*/false, /*reuse_b=*/false);
}

// Gq layout: 10 gates x [g00, g01, g10, g11]
__device__ __forceinline__ void set_gates_h(float* Gq, int lane) {
  if (lane < NQ) {
    const float r = 0.70710678118654752f;
    Gq[lane * 4 + 0] = r;
    Gq[lane * 4 + 1] = r;
    Gq[lane * 4 + 2] = r;
    Gq[lane * 4 + 3] = -r;
  }
}

__device__ __forceinline__ void set_gates_ry(float* Gq, const float* ang,
                                             float sign, int lane) {
  if (lane < NQ) {
    float s, c;
    __sincosf(0.5f * sign * ang[lane], &s, &c);
    Gq[lane * 4 + 0] = c;
    Gq[lane * 4 + 1] = -s;
    Gq[lane * 4 + 2] = s;
    Gq[lane * 4 + 3] = c;
  }
}

// M[r*16+c] = prod_{q=0..3} g_q[bit_q(r)][bit_q(c)]  (kron of 4 2x2 gates).
// Row r = (lane>>4) + 2*t, so bits 1..3 of r are the compile-time unroll
// index t -> every factor is a single 2-way select on a column bit.
__device__ __forceinline__ void build_M16(float* M, const float* g, int lane) {
  const float a00 = g[0],  a01 = g[1],  a10 = g[2],  a11 = g[3];
  const float b00 = g[4],  b01 = g[5],  b10 = g[6],  b11 = g[7];
  const float c00 = g[8],  c01 = g[9],  c10 = g[10], c11 = g[11];
  const float d00 = g[12], d01 = g[13], d10 = g[14], d11 = g[15];
  const int col = lane & 15;
  const int r0  = (lane >> 4) & 1;
  const int cb0 = col & 1, cb1 = (col >> 1) & 1;
  const int cb2 = (col >> 2) & 1, cb3 = (col >> 3) & 1;
  const float f0 = r0 ? (cb0 ? a11 : a10) : (cb0 ? a01 : a00);
#pragma unroll
  for (int t = 0; t < 8; ++t) {
    const float f1 = (t & 1) ? (cb1 ? b11 : b10) : (cb1 ? b01 : b00);
    const float f2 = (t & 2) ? (cb2 ? c11 : c10) : (cb2 ? c01 : c00);
    const float f3 = (t & 4) ? (cb3 ? d11 : d10) : (cb3 ? d01 : d00);
    M[lane + 32 * t] = f0 * f1 * f2 * f3;
  }
}

// Qubits 0..3: state as (64 rows o) x (16 cols); new = old * M^T.
// B fragment (M^T) loaded once; per chunk both planes' A-frags are preloaded
// and the two accumulation chains interleaved.
__device__ __forceinline__ void apply_lo2(float* Sre, float* Sim,
                                          const float* M, int lane) {
  const int lm16 = lane & 15;
  const int lhi  = lane >> 4;
  v2f b[4];
#pragma unroll
  for (int s = 0; s < 4; ++s) {
    const int k0 = 4 * s + 2 * lhi;
    b[s].x = M[lm16 * 16 + k0];       // B[k][n] = M[n*16+k], n = lane%16
    b[s].y = M[lm16 * 16 + k0 + 1];
  }
#pragma unroll
  for (int t = 0; t < 4; ++t) {
    const int row = t * 16 + lm16;    // A-frag row
    v2f aR[4], aI[4];
#pragma unroll
    for (int s = 0; s < 4; ++s) {
      const int k0 = 4 * s + 2 * lhi;
      aR[s].x = Sre[row * 16 + k0];
      aR[s].y = Sre[row * 16 + k0 + 1];
      aI[s].x = Sim[row * 16 + k0];
      aI[s].y = Sim[row * 16 + k0 + 1];
    }
    v8f accR = {}, accI = {};
#pragma unroll
    for (int s = 0; s < 4; ++s) {
      accR = wmma4(aR[s], b[s], accR);
      accI = wmma4(aI[s], b[s], accI);
    }
#pragma unroll
    for (int v = 0; v < 8; ++v) {     // D(m,n): lane = n+16*(m/8), vgpr = m%8
      Sre[(t * 16 + v + 8 * lhi) * 16 + lm16] = accR[v];
      Sim[(t * 16 + v + 8 * lhi) * 16 + lm16] = accI[v];
    }
  }
}

// Qubits 4..7: i = o*256 + m*16 + inn ; per o: new = M * X.
// A fragment (M) loaded once; per o-block both planes' B-frags preloaded and
// the two WMMA chains interleaved.
__device__ __forceinline__ void apply_mid2(float* Sre, float* Sim,
                                           const float* M, int lane) {
  const int lm16 = lane & 15;
  const int lhi  = lane >> 4;
  v2f a[4];
#pragma unroll
  for (int s = 0; s < 4; ++s) {
    const int k0 = 4 * s + 2 * lhi;
    a[s].x = M[lm16 * 16 + k0];       // A = M, row = lane%16
    a[s].y = M[lm16 * 16 + k0 + 1];
  }
#pragma unroll
  for (int o = 0; o < 4; ++o) {
    float* SR = Sre + o * 256;
    float* SI = Sim + o * 256;
    v2f bR[4], bI[4];
#pragma unroll
    for (int s = 0; s < 4; ++s) {     // preload all B-frags: 8 loads in flight
      const int k0 = 4 * s + 2 * lhi;
      bR[s].x = SR[k0 * 16 + lm16];
      bR[s].y = SR[(k0 + 1) * 16 + lm16];
      bI[s].x = SI[k0 * 16 + lm16];
      bI[s].y = SI[(k0 + 1) * 16 + lm16];
    }
    v8f accR = {}, accI = {};
#pragma unroll
    for (int s = 0; s < 4; ++s) {
      accR = wmma4(a[s], bR[s], accR);
      accI = wmma4(a[s], bI[s], accI);
    }
#pragma unroll
    for (int v = 0; v < 8; ++v) {
      SR[(v + 8 * lhi) * 16 + lm16] = accR[v];
      SI[(v + 8 * lhi) * 16 + lm16] = accI[v];
    }
  }
}

// Qubits 8,9: i = m*256 + inn ; M4 = G9 kron G8 applied along m (VALU).
__device__ __forceinline__ void apply_hi(float* Sre, float* Sim,
                                         const float* Gq, int lane) {
  const float e00 = Gq[32], e01 = Gq[33], e10 = Gq[34], e11 = Gq[35]; // g8
  const float f00 = Gq[36], f01 = Gq[37], f10 = Gq[38], f11 = Gq[39]; // g9
  float M4[4][4];
  M4[0][0] = f00 * e00; M4[0][1] = f00 * e01; M4[0][2] = f01 * e00; M4[0][3] = f01 * e01;
  M4[1][0] = f00 * e10; M4[1][1] = f00 * e11; M4[1][2] = f01 * e10; M4[1][3] = f01 * e11;
  M4[2][0] = f10 * e00; M4[2][1] = f10 * e01; M4[2][2] = f11 * e00; M4[2][3] = f11 * e01;
  M4[3][0] = f10 * e10; M4[3][1] = f10 * e11; M4[3][2] = f11 * e10; M4[3][3] = f11 * e11;
#pragma unroll
  for (int t = 0; t < 8; ++t) {
    const int inn = lane + 32 * t;
    float re[4], im[4], orr[4], oii[4];
#pragma unroll
    for (int m = 0; m < 4; ++m) {
      re[m] = Sre[m * 256 + inn];
      im[m] = Sim[m * 256 + inn];
    }
#pragma unroll
    for (int m = 0; m < 4; ++m) {
      float ar = 0.f, ai = 0.f;
#pragma unroll
      for (int n = 0; n < 4; ++n) { ar += M4[m][n] * re[n]; ai += M4[m][n] * im[n]; }
      orr[m] = ar; oii[m] = ai;
    }
#pragma unroll
    for (int m = 0; m < 4; ++m) {
      Sre[m * 256 + inn] = orr[m];
      Sim[m * 256 + inn] = oii[m];
    }
  }
}

// Full 10-qubit separable gate layer from LDS gate table Gq[40].
__device__ __forceinline__ void gate_layer(float* Sre, float* Sim, float* M,
                                           const float* Gq, int lane) {
  build_M16(M, Gq, lane);           // qubits 0..3
  __syncthreads();
  apply_lo2(Sre, Sim, M, lane);
  __syncthreads();
  build_M16(M, Gq + 16, lane);      // qubits 4..7
  __syncthreads();
  apply_mid2(Sre, Sim, M, lane);
  __syncthreads();
  apply_hi(Sre, Sim, Gq, lane);     // qubits 8,9
  __syncthreads();
}

// rz diagonal: phase(i) = sign * sum_q (b_q - 1/2) * x[(start + 9 - q) % 10]
// i = lane + 32*t : bits 0-4 from lane (precomputed), bits 5-9 from t.
__device__ __forceinline__ void diag_rz(float* Sre, float* Sim, const float* xs,
                                        int start, float sign, int lane) {
  float hq[NQ];
#pragma unroll
  for (int q = 0; q < NQ; ++q)
    hq[q] = 0.5f * sign * xs[(start + NQ - 1 - q) % DFEAT];
  float plo = 0.f;
#pragma unroll
  for (int q = 0; q < 5; ++q)
    plo += ((lane >> q) & 1) ? hq[q] : -hq[q];
  for (int t = 0; t < 32; ++t) {
    const int i = lane + 32 * t;
    float ph = plo;
#pragma unroll
    for (int q = 5; q < NQ; ++q)
      ph += ((t >> (q - 5)) & 1) ? hq[q] : -hq[q];
    float s, c;
    __sincosf(ph, &s, &c);
    const float r = Sre[i], m = Sim[i];
    Sre[i] = r * c - m * s;
    Sim[i] = r * s + m * c;
  }
}

// ring diagonal: edges (0,1)..(8,9),(9,0); phase += b_ctrl*(b_tgt-1/2)*ang[e]
__device__ __forceinline__ void diag_ring(float* Sre, float* Sim,
                                          const float* ang, float sign, int lane) {
  float h[NQ];
#pragma unroll
  for (int e = 0; e < NQ; ++e) h[e] = 0.5f * sign * ang[e];
  float plo = 0.f;
#pragma unroll
  for (int e = 0; e < 4; ++e) {                 // edges fully in lane bits
    const float tm = ((lane >> (e + 1)) & 1) ? h[e] : -h[e];
    plo += ((lane >> e) & 1) ? tm : 0.f;
  }
  const int   b4 = (lane >> 4) & 1;             // ctrl of edge 4
  const float t9 = (lane & 1) ? h[9] : -h[9];   // tgt factor of edge 9
  for (int t = 0; t < 32; ++t) {
    const int i = lane + 32 * t;
    float ph = plo;
    {                                           // edge 4: ctrl b4, tgt = t bit0
      const float tm = (t & 1) ? h[4] : -h[4];
      ph += b4 ? tm : 0.f;
    }
#pragma unroll
    for (int e = 5; e < 9; ++e) {               // edges fully in t bits
      const float tm = ((t >> (e - 4)) & 1) ? h[e] : -h[e];
      ph += ((t >> (e - 5)) & 1) ? tm : 0.f;
    }
    ph += ((t >> 4) & 1) ? t9 : 0.f;            // edge 9: ctrl = t bit4
    float s, c;
    __sincosf(ph, &s, &c);
    const float r = Sre[i], m = Sim[i];
    Sre[i] = r * c - m * s;
    Sim[i] = r * s + m * c;
  }
}

__global__ __launch_bounds__(32)
void qkta_pair_kernel(const float* __restrict__ data,
                      const float* __restrict__ params,
                      float* __restrict__ kout) {
  __shared__ float Sre[QDIM];
  __shared__ float Sim[QDIM];
  __shared__ float Mg[256];
  __shared__ float Gq[4 * NQ];        // current layer 2x2 gate entries
  __shared__ float Xf[2 * DFEAT];     // x1 | x2
  __shared__ float Pp[NLAY * 2 * NQ]; // all params

  const int lane = threadIdx.x;
  const int pair = blockIdx.x;
  const int i1 = pair >> 6;           // x1 = repeat(data, 64) -> row p/64
  const int i2 = pair & 63;           // x2 = tile(data, 64)   -> row p%64

#pragma unroll
  for (int t = 0; t < 4; ++t) {
    const int idx = lane + 32 * t;
    if (idx < NLAY * 2 * NQ) Pp[idx] = params[idx];
  }
  if (lane < DFEAT)            Xf[lane] = data[i1 * DFEAT + lane];
  else if (lane < 2 * DFEAT)   Xf[lane] = data[i2 * DFEAT + (lane - DFEAT)];

  for (int t = 0; t < 32; ++t) {
    const int i = lane + 32 * t;
    Sre[i] = (i == 0) ? 1.f : 0.f;
    Sim[i] = 0.f;
  }
  __syncthreads();

  // ---- forward: block_f(state, x1, params[j], j*NQ) ----
  for (int j = 0; j < NLAY; ++j) {
    set_gates_h(Gq, lane);
    __syncthreads();
    gate_layer(Sre, Sim, Mg, Gq, lane);              // H on all qubits
    diag_rz(Sre, Sim, Xf, j * NQ, 1.f, lane);        // rz_diag(x1, start)
    __syncthreads();
    set_gates_ry(Gq, Pp + (j * 2) * NQ, 1.f, lane);
    __syncthreads();
    gate_layer(Sre, Sim, Mg, Gq, lane);              // RY layer
    diag_ring(Sre, Sim, Pp + (j * 2 + 1) * NQ, 1.f, lane);
    __syncthreads();
  }

  // ---- adjoint: adj_block_f(state, x2, params[4-k], (NQ-k-1)*NQ) ----
  for (int k = 0; k < NLAY; ++k) {
    const int j = NLAY - 1 - k;
    diag_ring(Sre, Sim, Pp + (j * 2 + 1) * NQ, -1.f, lane);  // conj ring
    __syncthreads();
    set_gates_ry(Gq, Pp + (j * 2) * NQ, -1.f, lane);
    __syncthreads();
    gate_layer(Sre, Sim, Mg, Gq, lane);              // RY(-theta)
    diag_rz(Sre, Sim, Xf + DFEAT, (NQ - k - 1) * NQ, -1.f, lane); // conj rz
    __syncthreads();
    set_gates_h(Gq, lane);
    __syncthreads();
    gate_layer(Sre, Sim, Mg, Gq, lane);              // H (self-adjoint)
  }

  __syncthreads();
  if (lane == 0) {
    const float a = Sre[0], b = Sim[0];
    kout[pair] = a * a + b * b;                      // |<0|U2^ U1|0>|^2
  }
}

__global__ __launch_bounds__(256)
void qkta_reduce_kernel(const float* __restrict__ kvals,
                        const float* __restrict__ labels,
                        float* __restrict__ out) {
  __shared__ float skp[256];
  __shared__ float skk[256];
  const int t = threadIdx.x;
  float kp = 0.f, kk = 0.f;
  for (int p = t; p < NPTS * NPTS; p += 256) {
    const float kv = kvals[p];
    const float lm = labels[p >> 6] * labels[p & 63];
    kp += lm * kv;
    kk += kv * kv;
  }
  skp[t] = kp;
  skk[t] = kk;
  __syncthreads();
  for (int s = 128; s > 0; s >>= 1) {
    if (t < s) { skp[t] += skp[t + s]; skk[t] += skk[t + s]; }
    __syncthreads();
  }
  if (t == 0) {
    float sl = 0.f;
    for (int i = 0; i < NPTS; ++i) { const float l = labels[i]; sl += l * l; }
    const float slm = sl * sl;                 // sum(lm*lm) = (sum l^2)^2
    out[0] = skp[0] / sqrtf(skk[0] * slm);
  }
}

extern "C" void kernel_launch(void* const* d_in, const int* in_sizes, int n_in,
                              void* d_out, int out_size, void* d_ws, size_t ws_size,
                              hipStream_t stream) {
  const float* data   = (const float*)d_in[0];   // (64, 10)
  const float* labels = (const float*)d_in[1];   // (64,)
  const float* params = (const float*)d_in[2];   // (5, 2, 10)
  float* kvals = (float*)d_ws;                   // 4096 floats of scratch

  qkta_pair_kernel<<<NPTS * NPTS, 32, 0, stream>>>(data, params, kvals);
  qkta_reduce_kernel<<<1, 256, 0, stream>>>(kvals, labels, (float*)d_out);
}